// LinearizedContext_33131377721753
// MI455X (gfx1250) — compile-verified
//
#include <hip/hip_runtime.h>
#include <math.h>

// Problem constants (from reference): N=512 boxes, NC=151 classes, H=256 feat dim.
#define NBOX 512
#define NCLS 151
#define HDIM 256
#define PROB_N (NBOX * NCLS)   // 77312

typedef float v2f __attribute__((ext_vector_type(2)));
typedef float v8f __attribute__((ext_vector_type(8)));

// =====================================================================
// Kernel A: obj_dists2 = obj_feats @ W^T + b + obj_dists1   (f32 WMMA)
// One wave (32 lanes) computes one 16x16 output tile using
// V_WMMA_F32_16X16X4_F32, K swept 0..255 in steps of 4 (64 wmma issues).
//
// ISA lane layouts (CDNA5 ISA):
//   A 16x4 f32 : lanes 0-15 -> M=lane, VGPR0=K0,VGPR1=K1;
//                lanes 16-31 -> M=lane-16, VGPR0=K2,VGPR1=K3.
//   B 4x16 f32 : lanes 0-15 -> N=lane, VGPR0=K0 row, VGPR1=K1 row;
//                lanes 16-31 -> N=lane-16, VGPR0=K2 row, VGPR1=K3 row.
//   C/D 16x16  : lanes 0-15 N=lane M=v; lanes 16-31 N=lane-16 M=v+8.
//
// Tail handling: lanes whose class column >= 151 clamp their B pointer
// to W row 0 (always-legal load) and are simply never stored. WMMA
// columns are independent, so the garbage never contaminates valid
// outputs -- this keeps the inner loop branch-free and EXEC untouched
// (WMMA requires EXEC all-1s anyway).
// =====================================================================
__global__ __launch_bounds__(32)
void linear_wmma_kernel(const float* __restrict__ feats,   // [512,256]
                        const float* __restrict__ W,       // [151,256]
                        const float* __restrict__ bias,    // [151]
                        const float* __restrict__ d1,      // [512,151]
                        float* __restrict__ out)           // [512,151]
{
    const int lane = threadIdx.x;        // 0..31
    const int tile = blockIdx.x;         // 32 M-tiles * 10 N-tiles = 320
    const int tm   = tile / 10;
    const int tn   = tile % 10;
    const int half = lane >> 4;          // 0 or 1
    const int l15  = lane & 15;

    const int arow  = tm * 16 + l15;     // always < 512
    const int bcol  = tn * 16 + l15;     // may exceed 150 in last tile
    const bool bval = (bcol < NCLS);

    const float* aptr = feats + (size_t)arow * HDIM + 2 * half;
    const float* bptr = W + (size_t)(bval ? bcol : 0) * HDIM + 2 * half;

    v8f c = {};
#pragma unroll 8
    for (int k = 0; k < HDIM; k += 4) {
        // 8-byte aligned contiguous pair loads; both unconditional.
        const v2f a  = *(const v2f*)(aptr + k);
        const v2f bv = *(const v2f*)(bptr + k);
        // D = A(16x4)*B(4x16)+C ; args: neg_a,A,neg_b,B,c_mod,C,reuse_a,reuse_b
        c = __builtin_amdgcn_wmma_f32_16x16x4_f32(
                false, a, false, bv, (short)0, c, false, false);
    }

    if (bval) {
        const float bb = bias[bcol];
#pragma unroll
        for (int v = 0; v < 8; ++v) {
            const int mrow = tm * 16 + v + half * 8;
            const size_t o = (size_t)mrow * NCLS + bcol;
            out[o] = c[v] + bb + d1[o];
        }
    }
}

// =====================================================================
// Kernel B: row softmax over 151 classes, then probs[:,0] = 0.
// One wave per row; shfl_xor reductions (wave32).
// =====================================================================
__global__ __launch_bounds__(32)
void softmax_kernel(const float* __restrict__ dists,  // [512,151]
                    float* __restrict__ probs)        // [512,151]
{
    const int row  = blockIdx.x;
    const int lane = threadIdx.x;
    const float* rp = dists + (size_t)row * NCLS;

    float m = -3.0e38f;
    for (int c = lane; c < NCLS; c += 32) m = fmaxf(m, rp[c]);
    for (int off = 16; off > 0; off >>= 1) m = fmaxf(m, __shfl_xor(m, off, 32));

    float s = 0.0f;
    for (int c = lane; c < NCLS; c += 32) s += expf(rp[c] - m);
    for (int off = 16; off > 0; off >>= 1) s += __shfl_xor(s, off, 32);
    const float inv = 1.0f / s;

    float* op = probs + (size_t)row * NCLS;
    for (int c = lane; c < NCLS; c += 32)
        op[c] = (c == 0) ? 0.0f : expf(rp[c] - m) * inv;
}

// =====================================================================
// Kernel C: sequential greedy NMS. Single workgroup (1024 threads =
// 32 wave32 waves). The full 512x151 probs matrix (309 KB) lives in
// CDNA5's 320 KB LDS (impossible on CDNA4's 64 KB/CU). IoU is computed
// on demand (512 pairs/iter) instead of materializing the 39.6M-element
// overlap tensor (saves ~160 MB of traffic and 151x the IoU math).
// Argmax uses first-occurrence tie-break to match jnp.argmax.
// =====================================================================
__device__ __forceinline__ void argmax_comb(float& bv, int& bi, float ov, int oi)
{
    if (ov > bv || (ov == bv && oi < bi)) { bv = ov; bi = oi; }
}

__global__ __launch_bounds__(1024)
void nms_kernel(const float* __restrict__ probs_g,   // [512,151]
                const float* __restrict__ boxes,     // [512,151,4]
                float* __restrict__ preds_out)       // [512] (as float)
{
    extern __shared__ unsigned char smem[];
    float* probs = (float*)smem;                              // 77312 floats
    float* swv   = (float*)(smem + (size_t)PROB_N * 4);       // 32 floats
    int*   swi   = (int*)(swv + 32);                          // 32 ints
    int*   preds = swi + 32;                                  // 512 ints
    int*   sbc   = preds + NBOX;                              // 2 ints

    const int tid  = threadIdx.x;
    const int lane = tid & 31;
    const int wave = tid >> 5;

    for (int i = tid; i < PROB_N; i += 1024) probs[i] = probs_g[i];
    if (tid < NBOX) preds[tid] = 0;
    __syncthreads();

    for (int iter = 0; iter < NBOX; ++iter) {
        // ---- phase 1: global argmax over LDS-resident probs ----
        float bv = -3.0e38f;
        int   bi = PROB_N;
        for (int i = tid; i < PROB_N; i += 1024) {
            const float v = probs[i];
            if (v > bv) { bv = v; bi = i; }   // first occurrence within stride
        }
        for (int off = 16; off > 0; off >>= 1) {
            const float ov = __shfl_down(bv, off, 32);
            const int   oi = __shfl_down(bi, off, 32);
            argmax_comb(bv, bi, ov, oi);
        }
        if (lane == 0) { swv[wave] = bv; swi[wave] = bi; }
        __syncthreads();

        if (wave == 0) {
            float v2 = swv[lane];
            int   i2 = swi[lane];
            for (int off = 16; off > 0; off >>= 1) {
                const float ov = __shfl_down(v2, off, 32);
                const int   oi = __shfl_down(i2, off, 32);
                argmax_comb(v2, i2, ov, oi);
            }
            if (lane == 0) {
                const int box = i2 / NCLS;
                const int cls = i2 - box * NCLS;
                sbc[0] = box;
                sbc[1] = cls;
                preds[box] = cls;
            }
        }
        __syncthreads();
        const int box = sbc[0];
        const int cls = sbc[1];

        // ---- phase 2: zero column cls where IoU(box, j) >= 0.5 ----
        if (tid < NBOX) {
            const float4 A  = *(const float4*)(boxes + ((size_t)box * NCLS + cls) * 4);
            const float4 Bx = *(const float4*)(boxes + ((size_t)tid * NCLS + cls) * 4);
            float w = fminf(A.z, Bx.z) - fmaxf(A.x, Bx.x) + 1.0f;
            float h = fminf(A.w, Bx.w) - fmaxf(A.y, Bx.y) + 1.0f;
            w = fmaxf(w, 0.0f);
            h = fmaxf(h, 0.0f);
            const float inter = w * h;
            const float aA = (A.z - A.x + 1.0f) * (A.w - A.y + 1.0f);
            const float aB = (Bx.z - Bx.x + 1.0f) * (Bx.w - Bx.y + 1.0f);
            const float iou = inter / (aA + aB - inter);
            if (iou >= 0.5f) probs[(size_t)tid * NCLS + cls] = 0.0f;
        }
        __syncthreads();

        // ---- phase 3: kill selected row (after column suppression) ----
        if (tid < NCLS) probs[(size_t)box * NCLS + tid] = -1.0f;
        __syncthreads();
    }

    if (tid < NBOX) preds_out[tid] = (float)preds[tid];
}

// =====================================================================
// Launch: inputs are (obj_dists1, obj_feats, boxes_per_cls, W, b).
// d_out: [0,77312) = obj_dists2 (f32), [77312,77824) = obj_preds (as f32).
// d_ws : probs scratch (77312 floats).
// =====================================================================
extern "C" void kernel_launch(void* const* d_in, const int* in_sizes, int n_in,
                              void* d_out, int out_size, void* d_ws, size_t ws_size,
                              hipStream_t stream)
{
    (void)in_sizes; (void)n_in; (void)out_size; (void)ws_size;

    const float* obj_dists1 = (const float*)d_in[0];   // [512,151]
    const float* obj_feats  = (const float*)d_in[1];   // [512,256]
    const float* boxes      = (const float*)d_in[2];   // [512,151,4]
    const float* W          = (const float*)d_in[3];   // [151,256]
    const float* b          = (const float*)d_in[4];   // [151]

    float* out   = (float*)d_out;
    float* probs = (float*)d_ws;

    linear_wmma_kernel<<<dim3(32 * 10), dim3(32), 0, stream>>>(
        obj_feats, W, b, obj_dists1, out);

    softmax_kernel<<<dim3(NBOX), dim3(32), 0, stream>>>(out, probs);

    const size_t lds_bytes = (size_t)PROB_N * 4   // probs
                           + 32 * 4 + 32 * 4      // wave reduction scratch
                           + NBOX * 4             // preds
                           + 2 * 4;               // broadcast box/cls
    nms_kernel<<<dim3(1), dim3(1024), lds_bytes, stream>>>(
        probs, boxes, out + PROB_N);
}